// LSTMCRFTriggerTagger_13194139533460
// MI455X (gfx1250) — compile-verified
//
#include <hip/hip_runtime.h>
#include <hip/hip_bf16.h>

// Problem constants (from reference)
#define V_  50000
#define E_  64
#define H_  64
#define G_  256   // 4*H
#define B_  256
#define T_  512
#define K_  3

typedef __attribute__((ext_vector_type(2))) float v2f;
typedef __attribute__((ext_vector_type(8))) float v8f;

#define BT      16    // batch rows per workgroup
#define NWAVES  16    // one 16-col N-tile per wave -> 16*16 = 256 gate cols
#define NTHREADS (NWAVES * 32)

// ---------------------------------------------------------------------------
// Fused embedding-gather + input-projection + LSTM recurrence.
// One (direction, batch-tile) per workgroup; full T=512 scan stays on-chip.
//   A (16x128) = [ emb(x[:,t]) | h_prev ]  in LDS
//   B (128x256) = [ w_ih ; w_hh ]          register-resident (fully unrolled)
//   g = A*B + bias via v_wmma_f32_16x16x4_f32  (fp32-exact recurrence)
// ---------------------------------------------------------------------------
__global__ __launch_bounds__(NTHREADS)
void lstm_scan_kernel(const int* __restrict__ x, const float* __restrict__ emb,
                      const float* __restrict__ w_ih_f, const float* __restrict__ w_hh_f,
                      const float* __restrict__ b_ih_f, const float* __restrict__ b_hh_f,
                      const float* __restrict__ w_ih_b, const float* __restrict__ w_hh_b,
                      const float* __restrict__ b_ih_b, const float* __restrict__ b_hh_b,
                      float* __restrict__ hs_f, float* __restrict__ hs_b)
{
    __shared__ float lds_a[16][132];   // [row][k] : k<64 emb, k>=64 h   (pad 128->132)
    __shared__ float lds_g[16][260];   // gate pre-activations           (pad 256->260)
    __shared__ float lds_c[16][64];    // cell state

    const int tid   = threadIdx.x;
    const int lane  = tid & 31;
    const int wave  = tid >> 5;        // 0..15 == N-tile index == row index
    const int lhalf = lane >> 4;       // 0 | 1
    const int lmod  = lane & 15;

    const int dir   = blockIdx.x & 1;          // 0 = forward, 1 = backward
    const int b0    = (blockIdx.x >> 1) * BT;  // batch-tile origin

    const float* w_ih = dir ? w_ih_b : w_ih_f;
    const float* w_hh = dir ? w_hh_b : w_hh_f;
    const float* b_ih = dir ? b_ih_b : b_ih_f;
    const float* b_hh = dir ? b_hh_b : b_hh_f;
    float*       hs   = dir ? hs_b   : hs_f;

    // ---- preload this wave's weight B-fragments (K = 128 combined) --------
    // B[k][n]: k<64 -> w_ih[n][k], k>=64 -> w_hh[n][k-64]
    // ISA f32 B layout: lane -> N = lmod, K = ks*4 + 2*lhalf + {0,1}
    const int n = wave * 16 + lmod;            // this lane's output column
    const float bias = b_ih[n] + b_hh[n];
    v2f wf[32];
#pragma unroll
    for (int ks = 0; ks < 32; ++ks) {
        const int k = ks * 4 + lhalf * 2;
        v2f w;
        if (ks < 16) { w.x = w_ih[n * 64 + k];      w.y = w_ih[n * 64 + k + 1];      }
        else         { w.x = w_hh[n * 64 + k - 64]; w.y = w_hh[n * 64 + k - 63];     }
        wf[ks] = w;
    }

    // ---- init h = 0, c = 0 ----
    for (int i = tid; i < 16 * 64; i += NTHREADS) {
        const int r = i >> 6, cc = i & 63;
        lds_a[r][64 + cc] = 0.0f;
        lds_c[r][cc]      = 0.0f;
    }
    // ---- gather embeddings for step 0 (wave w loads row w, coalesced) ----
    {
        const int t0 = dir ? (T_ - 1) : 0;
        const int r  = wave;
        const int c2 = lane * 2;
        const int idx = x[(b0 + r) * T_ + t0];       // wave-uniform -> s_load
        const float* src = emb + (long)idx * E_ + c2;
        lds_a[r][c2 + 0] = src[0];
        lds_a[r][c2 + 1] = src[1];
    }
    __syncthreads();

    for (int s = 0; s < T_; ++s) {
        const int t = dir ? (T_ - 1 - s) : s;

        // ========== WMMA phase: g(:, wave-tile) = [emb|h] * B + bias ======
        v8f acc0, acc1;
#pragma unroll
        for (int v = 0; v < 8; ++v) { acc0[v] = bias; acc1[v] = 0.0f; }

#pragma unroll
        for (int ks = 0; ks < 16; ++ks) {
            // low-half K chain, k 0..63 (emb)
            {
                const int k = ks * 4 + lhalf * 2;
                v2f afrag;
                afrag.x = lds_a[lmod][k];
                afrag.y = lds_a[lmod][k + 1];
                acc0 = __builtin_amdgcn_wmma_f32_16x16x4_f32(
                    false, afrag, false, wf[ks], (short)0, acc0, false, false);
            }
            // high-half K chain, k 64..127 (h) (independent accumulator -> 2x ILP)
            {
                const int k = (ks + 16) * 4 + lhalf * 2;
                v2f afrag;
                afrag.x = lds_a[lmod][k];
                afrag.y = lds_a[lmod][k + 1];
                acc1 = __builtin_amdgcn_wmma_f32_16x16x4_f32(
                    false, afrag, false, wf[ks + 16], (short)0, acc1, false, false);
            }
        }
#pragma unroll
        for (int v = 0; v < 8; ++v) acc0[v] += acc1[v];

        // spill accumulator to LDS (C/D layout: row = v + 8*lhalf, col = lmod)
        {
            const int ncol = wave * 16 + lmod;
            const int rb   = 8 * lhalf;
#pragma unroll
            for (int v = 0; v < 8; ++v)
                lds_g[rb + v][ncol] = acc0[v];
        }
        __syncthreads();

        // ========== gate phase + next-step emb gather =====================
        {
            const int r   = wave;          // one batch row per wave
            const int hc0 = lane * 2;      // two h columns per lane
#pragma unroll
            for (int j = 0; j < 2; ++j) {
                const int hc = hc0 + j;
                const float gi = lds_g[r][hc];
                const float gf = lds_g[r][64 + hc];
                const float gg = lds_g[r][128 + hc];
                const float go = lds_g[r][192 + hc];
                const float si = 1.0f / (1.0f + __expf(-gi));
                const float sf = 1.0f / (1.0f + __expf(-gf));
                const float so = 1.0f / (1.0f + __expf(-go));
                const float c  = sf * lds_c[r][hc] + si * tanhf(gg);
                const float h  = so * tanhf(c);
                lds_c[r][hc]      = c;
                lds_a[r][64 + hc] = h;                         // feeds next step
                hs[((long)t * B_ + (b0 + r)) * H_ + hc] = h;   // stream out
            }
            if (s + 1 < T_) {
                const int t2  = dir ? (T_ - 2 - s) : (s + 1);
                const int idx = x[(b0 + r) * T_ + t2];         // wave-uniform
                const float* src = emb + (long)idx * E_ + hc0;
                lds_a[r][hc0 + 0] = src[0];
                lds_a[r][hc0 + 1] = src[1];
            }
        }
        __syncthreads();
    }
}

// ---------------------------------------------------------------------------
// Emissions: em[t,b,k] = fc_b[k] + sum_h hf*fc_w[k,h] + hb*fc_w[k,64+h]
// ---------------------------------------------------------------------------
__global__ __launch_bounds__(256)
void fc_kernel(const float* __restrict__ hs_f, const float* __restrict__ hs_b,
               const float* __restrict__ fc_w, const float* __restrict__ fc_b,
               float* __restrict__ em)
{
    const long i = (long)blockIdx.x * 256 + threadIdx.x;   // over T*B
    if (i >= (long)T_ * B_) return;
    const float* hf = hs_f + i * H_;
    const float* hb = hs_b + i * H_;
    float o0 = fc_b[0], o1 = fc_b[1], o2 = fc_b[2];
#pragma unroll 8
    for (int h = 0; h < 64; ++h) {
        const float v = hf[h];
        o0 += v * fc_w[0 * 128 + h];
        o1 += v * fc_w[1 * 128 + h];
        o2 += v * fc_w[2 * 128 + h];
    }
#pragma unroll 8
    for (int h = 0; h < 64; ++h) {
        const float v = hb[h];
        o0 += v * fc_w[0 * 128 + 64 + h];
        o1 += v * fc_w[1 * 128 + 64 + h];
        o2 += v * fc_w[2 * 128 + 64 + h];
    }
    em[i * K_ + 0] = o0; em[i * K_ + 1] = o1; em[i * K_ + 2] = o2;
}

// ---------------------------------------------------------------------------
// CRF NLL: one thread per batch row (B=256 = 1 block), K=3 forward algorithm.
// ---------------------------------------------------------------------------
__device__ __forceinline__ float lse3(float x0, float x1, float x2) {
    const float mx = fmaxf(x0, fmaxf(x1, x2));
    return mx + __logf(__expf(x0 - mx) + __expf(x1 - mx) + __expf(x2 - mx));
}

__global__ __launch_bounds__(256)
void crf_kernel(const int* __restrict__ y, const unsigned char* __restrict__ mask,
                const float* __restrict__ em, const float* __restrict__ start_t,
                const float* __restrict__ end_t, const float* __restrict__ trans,
                float* __restrict__ out)
{
    __shared__ float red[256];
    const int b = threadIdx.x;

    float st[3], et[3], tr[9];
#pragma unroll
    for (int k = 0; k < 3; ++k) { st[k] = start_t[k]; et[k] = end_t[k]; }
#pragma unroll
    for (int k = 0; k < 9; ++k) tr[k] = trans[k];

    int yprev = y[b * T_ + 0];
    const float* em0 = em + (long)b * K_;
    float score = st[yprev] + em0[yprev];
    float a0 = st[0] + em0[0];
    float a1 = st[1] + em0[1];
    float a2 = st[2] + em0[2];
    int cnt = mask[b * T_ + 0] ? 1 : 0;

    for (int t = 1; t < T_; ++t) {
        const int   yt = y[b * T_ + t];
        const float m  = mask[b * T_ + t] ? 1.0f : 0.0f;
        const float* emt = em + ((long)t * B_ + b) * K_;
        const float e0 = emt[0], e1 = emt[1], e2 = emt[2];
        const float ey = (yt == 0) ? e0 : ((yt == 1) ? e1 : e2);
        score += m * (tr[yprev * 3 + yt] + ey);

        const float n0 = lse3(a0 + tr[0], a1 + tr[3], a2 + tr[6]) + e0;
        const float n1 = lse3(a0 + tr[1], a1 + tr[4], a2 + tr[7]) + e1;
        const float n2 = lse3(a0 + tr[2], a1 + tr[5], a2 + tr[8]) + e2;
        if (m > 0.0f) { a0 = n0; a1 = n1; a2 = n2; cnt++; }
        yprev = yt;
    }

    const int   seq_end  = cnt - 1;
    const int   last_tag = y[b * T_ + seq_end];
    const float numerator = score + et[last_tag];
    const float logZ = lse3(a0 + et[0], a1 + et[1], a2 + et[2]);
    red[b] = numerator - logZ;
    __syncthreads();
#pragma unroll
    for (int s = 128; s > 0; s >>= 1) {
        if (b < s) red[b] += red[b + s];
        __syncthreads();
    }
    if (b == 0) out[0] = -red[0] / (float)B_;
}

// ---------------------------------------------------------------------------
extern "C" void kernel_launch(void* const* d_in, const int* in_sizes, int n_in,
                              void* d_out, int out_size, void* d_ws, size_t ws_size,
                              hipStream_t stream)
{
    const int*           x      = (const int*)d_in[0];
    const int*           y      = (const int*)d_in[1];
    const unsigned char* mask   = (const unsigned char*)d_in[2];
    const float*         emb    = (const float*)d_in[3];
    const float*         w_ih_f = (const float*)d_in[4];
    const float*         w_hh_f = (const float*)d_in[5];
    const float*         b_ih_f = (const float*)d_in[6];
    const float*         b_hh_f = (const float*)d_in[7];
    const float*         w_ih_b = (const float*)d_in[8];
    const float*         w_hh_b = (const float*)d_in[9];
    const float*         b_ih_b = (const float*)d_in[10];
    const float*         b_hh_b = (const float*)d_in[11];
    const float*         fc_w   = (const float*)d_in[12];
    const float*         fc_b   = (const float*)d_in[13];
    const float*         start_t= (const float*)d_in[14];
    const float*         end_t  = (const float*)d_in[15];
    const float*         trans  = (const float*)d_in[16];

    // workspace partition
    float* hs_f = (float*)d_ws;                      // T*B*H
    float* hs_b = hs_f + (long)T_ * B_ * H_;         // T*B*H
    float* em   = hs_b + (long)T_ * B_ * H_;         // T*B*K

    // 2 directions x (B/16) batch tiles, each scans all T on-chip
    lstm_scan_kernel<<<2 * (B_ / BT), NTHREADS, 0, stream>>>(
        x, emb, w_ih_f, w_hh_f, b_ih_f, b_hh_f,
        w_ih_b, w_hh_b, b_ih_b, b_hh_b, hs_f, hs_b);

    fc_kernel<<<(int)(((long)T_ * B_ + 255) / 256), 256, 0, stream>>>(
        hs_f, hs_b, fc_w, fc_b, em);

    crf_kernel<<<1, 256, 0, stream>>>(y, mask, em, start_t, end_t, trans,
                                      (float*)d_out);
}